// RecurrentNN_52252572123764
// MI455X (gfx1250) — compile-verified
//
#include <hip/hip_runtime.h>

typedef float v2f __attribute__((ext_vector_type(2)));
typedef float v8f __attribute__((ext_vector_type(8)));

#define IN_  2
#define HID_ 4
#define OUT_ 2
#define T_   1024

// ---- fast tanh: gfx1250 V_TANH_F32 (confirmed lowering in round 1). ----
#if __has_builtin(__builtin_amdgcn_tanhf)
__device__ __forceinline__ float fast_tanh(float x) { return __builtin_amdgcn_tanhf(x); }
#elif __has_builtin(__builtin_amdgcn_tanh_f32)
__device__ __forceinline__ float fast_tanh(float x) { return __builtin_amdgcn_tanh_f32(x); }
#else
__device__ __forceinline__ float fast_tanh(float x) {
  float e = __builtin_amdgcn_exp2f(x * 2.8853900817779268f);
  return 1.0f - 2.0f * __builtin_amdgcn_rcpf(e + 1.0f);
}
#endif

// Exchange low/high 16-lane halves of the wave (SWAPX16).
__device__ __forceinline__ float swap16(float v) {
  int i = __float_as_int(v);
  i = __builtin_amdgcn_ds_swizzle(i, 0x401F);
  return __int_as_float(i);
}

// One wave (32 lanes) advances 16 independent LSTM cells through all T steps.
// WMMA 16x16x4 f32 layout:
//   A (16x4): lane m holds A[m][2*hi], A[m][2*hi+1]        (hi = lane>=16)
//   B (4x16): lane n holds B[2*hi][n%16], B[2*hi+1][n%16]
//   C/D     : lane n, vgpr r holds D[r + 8*hi][n%16]
// Rows = gates [i(0-3), f(4-7), g(8-11), o(12-15)], cols = batch.
// Low lanes own i,f (sigmoid); high lanes own g (tanh), o (sigmoid).
__global__ __launch_bounds__(32) void lstm_wmma16x16x4_kernel(
    const float* __restrict__ x,     // [B, T, IN]
    const float* __restrict__ w_ih,  // [16, 2]
    const float* __restrict__ w_hh,  // [16, 4]
    const float* __restrict__ b_ih,  // [16]
    const float* __restrict__ b_hh,  // [16]
    const float* __restrict__ w_lin, // [2, 4]
    const float* __restrict__ b_lin, // [2]
    float* __restrict__ out,         // [B, T, OUT]
    int Btotal)
{
  const int lane = threadIdx.x & 31;
  const int m    = lane & 15;          // A row / B,C,D column index
  const bool hi  = lane >= 16;
  const int hk   = hi ? 2 : 0;         // K-pair owned by this half

  int batch = blockIdx.x * 16 + m;
  if (batch >= Btotal) batch = Btotal - 1;
  // Clamped lanes compute the identical sequence as lane (Btotal-1)%16 of this
  // block would, so their duplicate stores write identical values (benign).

  // A1 = w_hh (16x4); A2 = w_ih zero-padded to 16x4 (K=2,3 -> 0).
  v2f A1; A1.x = w_hh[m * HID_ + hk];  A1.y = w_hh[m * HID_ + hk + 1];
  v2f A2; A2.x = hi ? 0.0f : w_ih[m * IN_ + 0];
          A2.y = hi ? 0.0f : w_ih[m * IN_ + 1];

  // C = bias broadcast over batch columns.
  v8f Cb;
#pragma unroll
  for (int r = 0; r < 8; ++r) {
    const int row = r + (hi ? 8 : 0);
    Cb[r] = b_ih[row] + b_hh[row];
  }

  // Output projection: each half computes only its own component.
  float wsel[HID_], bsel;
  {
    const int orow = hi ? 1 : 0;
    bsel = b_lin[orow];
#pragma unroll
    for (int k = 0; k < HID_; ++k) wsel[k] = w_lin[orow * HID_ + k];
  }

  // Activation y = a * tanh(s * x) + b:
  //   sigmoid: (0.5, 0.5, 0.5);  tanh: (1, 1, 0).
  // regs 0..3: low=sigmoid(i), high=tanh(g) -> per-half consts.
  // regs 4..7: sigmoid on both halves (f / o) -> literals.
  const float s03 = hi ? 1.0f : 0.5f;
  const float a03 = hi ? 1.0f : 0.5f;
  const float b03 = hi ? 0.0f : 0.5f;

  const float* xp = x + (size_t)batch * T_ * IN_;
  // Split store: lane b -> out[b][t][0], lane b+16 -> out[b][t][1].
  float* op = out + (size_t)batch * T_ * OUT_ + (hi ? 1 : 0);

  float cst[HID_] = {0.f, 0.f, 0.f, 0.f};
  v2f Bh; Bh.x = 0.0f; Bh.y = 0.0f;    // H in B-operand layout: h[2*hi + {0,1}]

  // Software-pipelined x stream: xv holds x_t while x_{t+1} is in flight.
  v2f xv = *(const v2f*)(xp);

  for (int t = 0; t < T_; ++t) {
    // Issue next-step load early; clamp index so the last iteration re-reads t.
    const int tn = (t + 1 < T_) ? (t + 1) : t;
    const float* xnp = xp + tn * IN_;
    v2f xnext = *(const v2f*)(xnp);
    __builtin_prefetch((const char*)xnp + 256, 0, 3);  // global_prefetch_b8, OOB-safe

    v2f Bx; Bx.x = hi ? 0.0f : xv.x;  Bx.y = hi ? 0.0f : xv.y;

    // gates = W_ih*x + bias, then += W_hh*h : two chained WMMAs.
    v8f acc = __builtin_amdgcn_wmma_f32_16x16x4_f32(
        false, A2, false, Bx, (short)0, Cb, false, false);
    acc = __builtin_amdgcn_wmma_f32_16x16x4_f32(
        false, A1, false, Bh, (short)0, acc, false, false);

    // Activations (uniform code, per-half constants).
    float y[8];
#pragma unroll
    for (int r = 0; r < 4; ++r) y[r] = a03 * fast_tanh(s03 * acc[r]) + b03;
#pragma unroll
    for (int r = 4; r < 8; ++r) y[r] = 0.5f * fast_tanh(0.5f * acc[r]) + 0.5f;

    // Exchange halves so every lane sees all 16 gates of its batch column.
    float z[8];
#pragma unroll
    for (int r = 0; r < 8; ++r) z[r] = swap16(y[r]);

    // Cell update (redundant in both halves; both results are identical).
    float hv[HID_];
#pragma unroll
    for (int k = 0; k < HID_; ++k) {
      const float iv = hi ? z[k]     : y[k];
      const float fv = hi ? z[4 + k] : y[4 + k];
      const float gv = hi ? y[k]     : z[k];
      const float ov = hi ? y[4 + k] : z[4 + k];
      cst[k] = fv * cst[k] + iv * gv;
      hv[k]  = ov * fast_tanh(cst[k]);
    }

    // Rebuild H in B-operand layout (pure per-lane selects).
    Bh.x = hi ? hv[2] : hv[0];
    Bh.y = hi ? hv[3] : hv[1];

    // Output projection: each half computes + stores its own component.
    // Unpredicated store -> no exec-mask churn in the loop.
    float oval = bsel;
#pragma unroll
    for (int k = 0; k < HID_; ++k) oval += wsel[k] * hv[k];
    op[t * OUT_] = oval;

    xv = xnext;
  }
}

extern "C" void kernel_launch(void* const* d_in, const int* in_sizes, int n_in,
                              void* d_out, int out_size, void* d_ws, size_t ws_size,
                              hipStream_t stream) {
  const float* x     = (const float*)d_in[0];
  const float* w_ih  = (const float*)d_in[1];
  const float* w_hh  = (const float*)d_in[2];
  const float* b_ih  = (const float*)d_in[3];
  const float* b_hh  = (const float*)d_in[4];
  const float* w_lin = (const float*)d_in[5];
  const float* b_lin = (const float*)d_in[6];
  float* out = (float*)d_out;

  const int B = in_sizes[0] / (T_ * IN_);   // 8192
  const int nblocks = (B + 15) / 16;        // 512 one-wave workgroups
  hipLaunchKernelGGL(lstm_wmma16x16x4_kernel, dim3(nblocks), dim3(32), 0, stream,
                     x, w_ih, w_hh, b_ih, b_hh, w_lin, b_lin, out, B);
}